// ModelNew_14723147890950
// MI455X (gfx1250) — compile-verified
//
#include <hip/hip_runtime.h>

// C = tril(A * B), A,B fp32 4096x4096 lower-triangular.
// gfx1250 / MI455X: fp32 WMMA 16x16x4, wave32, 128x128 block tiles,
// double-buffered LDS staged via global_load_async_to_lds_b128 (ASYNCcnt).

#define NMAT 4096
#define BM 128
#define BN 128
#define BK 32
#define AS_STRIDE (BK + 4)   // 36 dwords: conflict-free A-fragment reads
#define BS_STRIDE (BN + 32)  // 160 dwords: lanes16-31 in other bank half

typedef __attribute__((ext_vector_type(2))) float v2f;
typedef __attribute__((ext_vector_type(4))) float v4f;
typedef __attribute__((ext_vector_type(8))) float v8f;

// GVS-mode async copy: copies 16B per lane from global memory at base plus voff into LDS at lds_off.
static __device__ __forceinline__ void async_load_b128(unsigned lds_off,
                                                       unsigned voff,
                                                       const float* __restrict__ base) {
    asm volatile("global_load_async_to_lds_b128 %0, %1, %2"
                 :: "v"(lds_off), "v"(voff), "s"(base)
                 : "memory");
}

static __device__ __forceinline__ void wait_async0() {
#if __has_builtin(__builtin_amdgcn_s_wait_asynccnt)
    __builtin_amdgcn_s_wait_asynccnt(0);
#else
    asm volatile("s_wait_asynccnt 0x0" ::: "memory");
#endif
}

__global__ __launch_bounds__(256)
void tril_mm_wmma_f32(const float* __restrict__ A,
                      const float* __restrict__ B,
                      float* __restrict__ C)
{
    const int bc = blockIdx.x;   // block column
    const int br = blockIdx.y;   // block row
    const int t  = threadIdx.x;

    // ---- Strictly-upper block: structurally zero, just fill (d_out poisoned).
    if (br < bc) {
        v4f z = {};
        #pragma unroll
        for (int p = 0; p < 16; ++p) {
            int s   = t + p * 256;
            int row = s >> 5;
            int c4  = (s & 31) << 2;
            *(v4f*)(C + (size_t)(br * BM + row) * NMAT + bc * BN + c4) = z;
        }
        return;
    }

    __shared__ __align__(16) float As[2][BM * AS_STRIDE];
    __shared__ __align__(16) float Bs[2][BK * BS_STRIDE];

    const int wave = t >> 5;
    const int lane = t & 31;
    const int half = lane >> 4;    // K-half select per ISA 7.12.2 fragment layout
    const int nl   = lane & 15;

    // Per-thread staging coordinates (byte offsets fit u32: matrices are 64MB).
    const int rowa = t >> 3;             // A tile: 0..31, +32 per pass
    const int c4a  = (t & 7) << 2;       // A tile k-subcolumn
    const int rk0  = t >> 5;             // B tile: 0..7, +8 per pass
    const int c4b  = (t & 31) << 2;      // B tile column

    const unsigned ldsA0 = (unsigned)(size_t)&As[0][0];
    const unsigned ldsA1 = (unsigned)(size_t)&As[1][0];
    const unsigned ldsB0 = (unsigned)(size_t)&Bs[0][0];
    const unsigned ldsB1 = (unsigned)(size_t)&Bs[1][0];

    v8f zacc = {};
    v8f acc[8];
    #pragma unroll
    for (int m = 0; m < 8; ++m) acc[m] = zacc;

    // Triangular k-range: B[k][j]!=0 needs k>=j ; A[i][k]!=0 needs k<=i.
    const int kStart = bc * BM;
    const int kEnd   = (br + 1) * BM;

    // ---- async stage of chunk `kb` into buffer selected by (la,lb) bases
    auto stage = [&](unsigned la, unsigned lb, int kb) {
        #pragma unroll
        for (int p = 0; p < 4; ++p) {
            int r = rowa + 32 * p;
            unsigned loff = la + (unsigned)((r * AS_STRIDE + c4a) * 4);
            unsigned goff = (unsigned)(((br * BM + r) * NMAT + kb + c4a) * 4);
            async_load_b128(loff, goff, A);
        }
        #pragma unroll
        for (int p = 0; p < 4; ++p) {
            int rk = rk0 + 8 * p;
            unsigned loff = lb + (unsigned)((rk * BS_STRIDE + c4b) * 4);
            unsigned goff = (unsigned)(((kb + rk) * NMAT + bc * BN + c4b) * 4);
            async_load_b128(loff, goff, B);
        }
    };

    // Prologue: fill buffer 0.
    stage(ldsA0, ldsB0, kStart);
    wait_async0();
    __syncthreads();

    int buf = 0;
    for (int kb = kStart; kb < kEnd; kb += BK) {
        // Prefetch next chunk into the other buffer (overlaps with WMMAs).
        const int knext = kb + BK;
        if (knext < kEnd)
            stage(buf ? ldsA0 : ldsA1, buf ? ldsB0 : ldsB1, knext);

        // ---- compute: 8 k-substeps x 8 row-tiles of V_WMMA_F32_16X16X4_F32
        const float* __restrict__ as = &As[buf][0];
        const float* __restrict__ bs = &Bs[buf][0];
        const int colbase = wave * 16;
        #pragma unroll
        for (int kk = 0; kk < BK; kk += 4) {
            const int k0 = kk + 2 * half;   // lanes16-31 hold K+2,K+3
            v2f bfrag;
            bfrag.x = bs[(k0    ) * BS_STRIDE + colbase + nl];
            bfrag.y = bs[(k0 + 1) * BS_STRIDE + colbase + nl];
            #pragma unroll
            for (int m = 0; m < 8; ++m) {
                v2f afrag = *(const v2f*)&as[(m * 16 + nl) * AS_STRIDE + k0];
                acc[m] = __builtin_amdgcn_wmma_f32_16x16x4_f32(
                    false, afrag, false, bfrag, (short)0, acc[m], false, false);
            }
        }

        wait_async0();       // next buffer fully resident (no-op on last chunk)
        __syncthreads();     // all waves done reading current buffer
        buf ^= 1;
    }

    // ---- epilogue: C/D layout -> VGPR r holds rows {m*16+r(+8 for hi lanes)}
    const int colg = bc * BN + wave * 16 + nl;
    #pragma unroll
    for (int m = 0; m < 8; ++m) {
        const int rowbase = br * BM + m * 16 + half * 8;
        #pragma unroll
        for (int r = 0; r < 8; ++r) {
            C[(size_t)(rowbase + r) * NMAT + colg] = acc[m][r];
        }
    }
}

extern "C" void kernel_launch(void* const* d_in, const int* in_sizes, int n_in,
                              void* d_out, int out_size, void* d_ws, size_t ws_size,
                              hipStream_t stream) {
    (void)in_sizes; (void)n_in; (void)d_ws; (void)ws_size; (void)out_size;
    const float* A = (const float*)d_in[0];
    const float* B = (const float*)d_in[1];
    float*       C = (float*)d_out;

    dim3 grid(NMAT / BN, NMAT / BM);   // 32x32 blocks; upper blocks zero-fill
    dim3 block(256);                   // 8 wave32s
    hipLaunchKernelGGL(tril_mm_wmma_f32, grid, block, 0, stream, A, B, C);
}